// LIFSpike_3599182594455
// MI455X (gfx1250) — compile-verified
//
#include <hip/hip_runtime.h>
#include <stdint.h>

#ifndef __has_builtin
#define __has_builtin(x) 0
#endif

typedef float v4f __attribute__((ext_vector_type(4)));
typedef int   v4i __attribute__((ext_vector_type(4)));

static constexpr float kVth   = 0.2f;
static constexpr float kTau   = 0.25f;   // 2^-2: exact multiply
static constexpr int   kT     = 8;       // timesteps, innermost/contiguous axis
static constexpr int   kBlk   = 256;     // 8 x wave32
static constexpr int   kTiles = 8;       // tiles per block (double-buffered pipeline)

// Sequential LIF scan over 8 timesteps held in registers.
// Matches reference bit-for-bit: o in {0,1} so tau*u*(1-o) == select(o, 0, tau*u),
// and tau is a power of two so fma(tau,u,x) rounds identically to (tau*u)+x.
__device__ __forceinline__ void lif_scan8(const float xin[kT], float o[kT]) {
  float u = 0.0f, sp = 0.0f;
#pragma unroll
  for (int t = 0; t < kT; ++t) {
    float keep = (sp == 0.0f) ? u : 0.0f;       // u * (1 - o_prev)
    u  = __builtin_fmaf(kTau, keep, xin[t]);    // tau*u*(1-o_prev) + x_t
    sp = (u > kVth) ? 1.0f : 0.0f;              // heaviside(u - Vth)
    o[t] = sp;
  }
}

#if defined(__HIP_DEVICE_COMPILE__)
#if __has_builtin(__builtin_amdgcn_global_load_async_to_lds_b128)
#define LIF_ASYNC 1
// Builtin signature (from clang diagnostic): arg0 = int4 addrspace(1)* (global src),
// arg1 = int4 addrspace(3)* (LDS dst), arg2/arg3 = imm offset / imm cpol.
typedef __attribute__((address_space(1))) v4i as1v4i;
typedef __attribute__((address_space(3))) v4i as3v4i;
__device__ __forceinline__ as1v4i* as1p(const void* p) {
  // global VA == generic VA for global memory
  return (as1v4i*)(uintptr_t)p;
}
__device__ __forceinline__ as3v4i* as3p(void* p) {
  // low 32 bits of a generic LDS address are the LDS byte offset (ISA 10.2)
  return (as3v4i*)(uint32_t)(uintptr_t)p;
}
#endif
#endif

__global__ __launch_bounds__(kBlk) void lif_pipe(const float* __restrict__ x,
                                                 float* __restrict__ out,
                                                 long long nLoc) {
#if defined(LIF_ASYNC)
  // ---- gfx1250 async-to-LDS double-buffered streaming pipeline ----
  __shared__ float lds[2][kBlk * kT];  // 2 x 8 KB tiles
  const int tid = threadIdx.x;
  const long long tile0    = (long long)blockIdx.x * kTiles;
  const long long waveBase = (long long)(threadIdx.x & ~31u);  // wave-uniform

  auto prefetch = [&](int t) {
    long long loc = (tile0 + t) * kBlk + tid;
    if (loc < nLoc) {
      as1v4i* g = as1p(x + loc * kT);                 // 32 B per lane
      as3v4i* l = as3p(&lds[t & 1][tid * kT]);        // lane-private LDS slice
      __builtin_amdgcn_global_load_async_to_lds_b128(g,     l,     0, 0);
      __builtin_amdgcn_global_load_async_to_lds_b128(g + 1, l + 1, 0, 0);
    }
  };

  prefetch(0);
#pragma unroll
  for (int t = 0; t < kTiles; ++t) {
    // Wave-uniform predicate: will the next prefetch issue for this wave?
    // (prefetch issues iff any lane of the wave is in range -> +2 on ASYNCcnt)
    bool more = (t + 1 < kTiles) &&
                (((tile0 + t + 1) * kBlk + waveBase) < nLoc);
    if (more) {
      prefetch(t + 1);                                // next tile in flight
#if __has_builtin(__builtin_amdgcn_s_wait_asynccnt)
      __builtin_amdgcn_s_wait_asynccnt(2);            // oldest 2 (tile t) retired
#else
      asm volatile("s_wait_asynccnt 2" ::: "memory");
#endif
    } else {
#if __has_builtin(__builtin_amdgcn_s_wait_asynccnt)
      __builtin_amdgcn_s_wait_asynccnt(0);
#else
      asm volatile("s_wait_asynccnt 0" ::: "memory");
#endif
    }
    long long loc = (tile0 + t) * kBlk + tid;
    if (loc < nLoc) {
      // Each lane reads back exactly the bytes it requested -> no barriers needed.
      const v4f* lp = (const v4f*)&lds[t & 1][tid * kT];
      v4f a = lp[0];
      v4f b = lp[1];
      float r[kT] = {a.x, a.y, a.z, a.w, b.x, b.y, b.z, b.w};
      float o[kT];
      lif_scan8(r, o);
      v4f oa = {o[0], o[1], o[2], o[3]};
      v4f ob = {o[4], o[5], o[6], o[7]};
      v4f* ov = (v4f*)(out + loc * kT);
      __builtin_nontemporal_store(oa, ov);            // single-touch: NT stores
      __builtin_nontemporal_store(ob, ov + 1);
    }
  }
#else
  // ---- fallback: straight b128 non-temporal streaming (same mapping) ----
  const int tid = threadIdx.x;
  const long long tile0 = (long long)blockIdx.x * kTiles;
#pragma unroll
  for (int t = 0; t < kTiles; ++t) {
    long long loc = (tile0 + t) * kBlk + tid;
    if (loc < nLoc) {
      const v4f* xv = (const v4f*)(x + loc * kT);
      v4f a = __builtin_nontemporal_load(xv);
      v4f b = __builtin_nontemporal_load(xv + 1);
      float r[kT] = {a.x, a.y, a.z, a.w, b.x, b.y, b.z, b.w};
      float o[kT];
      lif_scan8(r, o);
      v4f oa = {o[0], o[1], o[2], o[3]};
      v4f ob = {o[4], o[5], o[6], o[7]};
      v4f* ov = (v4f*)(out + loc * kT);
      __builtin_nontemporal_store(oa, ov);
      __builtin_nontemporal_store(ob, ov + 1);
    }
  }
#endif
}

extern "C" void kernel_launch(void* const* d_in, const int* in_sizes, int n_in,
                              void* d_out, int out_size, void* d_ws, size_t ws_size,
                              hipStream_t stream) {
  (void)n_in; (void)out_size; (void)d_ws; (void)ws_size;
  const float* x = (const float*)d_in[0];
  float* out = (float*)d_out;
  const long long n = (long long)in_sizes[0];     // 32*128*32*32*8
  const long long nLoc  = n / kT;                 // independent LIF chains
  const long long tiles = (nLoc + kBlk - 1) / kBlk;
  int blocks = (int)((tiles + kTiles - 1) / kTiles);
  if (blocks < 1) blocks = 1;
  lif_pipe<<<blocks, kBlk, 0, stream>>>(x, out, nLoc);
}